// EKCL_69449621176500
// MI455X (gfx1250) — compile-verified
//
#include <hip/hip_runtime.h>
#include <math.h>

// CDNA5 (gfx1250) wave32 WMMA kernel: Karcher mean on S^127, 8000 groups of 32x128.
// One wave = one group. dots GEMV via V_WMMA_F32_16X16X4_F32 (matrix pipe, 4-way
// accumulator ILP), transcendentals + exp-map on VALU (co-execution).
// Register budget kept to ~200 VGPRs (B-fragments only) for occupancy; the
// weighted column-sum streams the normalized tile from LDS instead.

typedef float v2f __attribute__((ext_vector_type(2)));
typedef float v4f __attribute__((ext_vector_type(4)));
typedef float v8f __attribute__((ext_vector_type(8)));

#define KN 32        // points per group
#define KD 128       // dimension
#define SXS 132      // padded LDS row stride (floats): conflict-free B-frag gather,
                     // keeps 16B slots aligned (132 % 4 == 0)
#define NITER 20
#define EPSF  1e-7f
#define CLIPF (1.0f - 1e-7f)

__global__ void __launch_bounds__(32) EKCL_karcher_zero(float* out, int n) {
  int i = threadIdx.x;
  if (i < n) out[i] = 0.0f;
}

__global__ void __launch_bounds__(32) EKCL_karcher_wmma(
    const float* __restrict__ X, float* __restrict__ out, float inv_groups)
{
  __shared__ float s_x[KN * SXS];    // normalized tile, padded row-major [n][d]
  __shared__ float s_red[KN * 33];   // padded transpose-reduce scratch
  __shared__ float s_mu[KD];         // current mean (A-fragment source)
  __shared__ float s_wd[2 * KN];     // {w_n, dot_n} broadcast pairs

  const int lane = threadIdx.x;
  const float* Xg = X + (size_t)blockIdx.x * (KN * KD);
  const float invN = 1.0f / 32.0f;

  // ---- Load (coalesced b128, non-temporal: zero reuse), stage raw, sq-partials ----
#pragma unroll
  for (int n = 0; n < KN; ++n) {
    const v4f v =
        __builtin_nontemporal_load((const v4f*)(Xg + n * KD + 4 * lane));
    *(v4f*)&s_x[n * SXS + 4 * lane] = v;
    s_red[n * 33 + lane] = v[0]*v[0] + v[1]*v[1] + v[2]*v[2] + v[3]*v[3];
  }
  __syncthreads();

  // ---- Row norms: lane sums row `lane` of the padded matrix (conflict-free) ----
  float rs = 0.0f;
#pragma unroll
  for (int i = 0; i < KN; ++i) rs += s_red[lane * 33 + i];
  const float rscale = 1.0f / fmaxf(sqrtf(rs), 1e-12f);
  s_red[lane] = rscale;              // broadcast slot (reads above already issued)
  __syncthreads();

  // ---- Normalize tile in LDS (lane-private column slots) + mu partials ----
  float mu0 = 0.f, mu1 = 0.f, mu2 = 0.f, mu3 = 0.f;
#pragma unroll
  for (int n = 0; n < KN; ++n) {
    const float sc = s_red[n];
    v4f v = *(const v4f*)&s_x[n * SXS + 4 * lane];
    v *= sc;
    *(v4f*)&s_x[n * SXS + 4 * lane] = v;
    mu0 += v[0]; mu1 += v[1]; mu2 += v[2]; mu3 += v[3];
  }
  __syncthreads();

  // ---- mu = normalize(mean); lane holds mu[4*lane..4*lane+3] ----
  mu0 *= invN; mu1 *= invN; mu2 *= invN; mu3 *= invN;
  {
    float mp = mu0*mu0 + mu1*mu1 + mu2*mu2 + mu3*mu3;
#pragma unroll
    for (int m = 16; m >= 1; m >>= 1) mp += __shfl_xor(mp, m, 32);
    const float msc = 1.0f / fmaxf(sqrtf(mp), 1e-12f);
    mu0 *= msc; mu1 *= msc; mu2 *= msc; mu3 *= msc;
  }
  *(v4f*)&s_mu[4 * lane] = (v4f){mu0, mu1, mu2, mu3};
  __syncthreads();

  // ---- WMMA B-fragments of Xn (16x16x4 f32 layout), register-resident.
  //      chunk c: VGPR0 = {K=0 | lanes 0-15, K=2 | lanes 16-31}, VGPR1 = {K=1, K=3}
  //      Padded stride makes this gather bank-conflict-free. ----
  const int r0   = lane & 15;
  const int r1   = r0 + 16;
  const int doff = (lane < 16) ? 0 : 2;
  v2f xb0[32], xb1[32];
#pragma unroll
  for (int c = 0; c < 32; ++c) {
    xb0[c] = *(const v2f*)&s_x[r0 * SXS + 4 * c + doff];
    xb1[c] = *(const v2f*)&s_x[r1 * SXS + 4 * c + doff];
  }

  const v8f zero8 = {0.f, 0.f, 0.f, 0.f, 0.f, 0.f, 0.f, 0.f};

  // ---- Karcher iterations ----
#pragma unroll 1
  for (int it = 0; it < NITER; ++it) {
    // dots[n] = <mu, Xn[n]> : A = mu replicated across rows; even/odd chunk
    // accumulators break the XDL RAW chain (only C[0] is consumed -> cheap merge).
    v8f a0e = zero8, a0o = zero8, a1e = zero8, a1o = zero8;
#pragma unroll
    for (int c = 0; c < 32; c += 2) {
      const v2f ae = *(const v2f*)&s_mu[4 * c + doff];
      const v2f ao = *(const v2f*)&s_mu[4 * (c + 1) + doff];
      a0e = __builtin_amdgcn_wmma_f32_16x16x4_f32(false, ae, false, xb0[c],
                                                  (short)0, a0e, false, false);
      a1e = __builtin_amdgcn_wmma_f32_16x16x4_f32(false, ae, false, xb1[c],
                                                  (short)0, a1e, false, false);
      a0o = __builtin_amdgcn_wmma_f32_16x16x4_f32(false, ao, false, xb0[c + 1],
                                                  (short)0, a0o, false, false);
      a1o = __builtin_amdgcn_wmma_f32_16x16x4_f32(false, ao, false, xb1[c + 1],
                                                  (short)0, a1o, false, false);
    }
    const float dot_raw =
        (lane < 16) ? (a0e[0] + a0o[0]) : (a1e[0] + a1o[0]);
    const float dcl   = fminf(fmaxf(dot_raw, -CLIPF), CLIPF);
    const float theta = acosf(dcl);
    const float w     = theta / fmaxf(sinf(theta), EPSF);
    *(v2f*)&s_wd[2 * lane] = (v2f){w, dot_raw};
    __syncthreads();

    // v = (1/N) * ( sum_n w_n x_n - (sum_n w_n dot_n) * mu ), lane-local in d;
    // x_n streamed from the LDS-resident normalized tile.
    float y0 = 0.f, y1 = 0.f, y2 = 0.f, y3 = 0.f, sacc = 0.f;
#pragma unroll
    for (int n = 0; n < KN; ++n) {
      const v2f wd = *(const v2f*)&s_wd[2 * n];                  // broadcast
      const v4f xv = *(const v4f*)&s_x[n * SXS + 4 * lane];
      y0 = fmaf(wd[0], xv[0], y0);
      y1 = fmaf(wd[0], xv[1], y1);
      y2 = fmaf(wd[0], xv[2], y2);
      y3 = fmaf(wd[0], xv[3], y3);
      sacc = fmaf(wd[0], wd[1], sacc);
    }
    const float v0 = (y0 - sacc * mu0) * invN;
    const float v1 = (y1 - sacc * mu1) * invN;
    const float v2 = (y2 - sacc * mu2) * invN;
    const float v3 = (y3 - sacc * mu3) * invN;

    float vp = v0*v0 + v1*v1 + v2*v2 + v3*v3;
#pragma unroll
    for (int m = 16; m >= 1; m >>= 1) vp += __shfl_xor(vp, m, 32);
    const float vn = fmaxf(sqrtf(vp), EPSF);     // vs_norm (STEP = 1)

    // exp map + renormalize
    const float cv = cosf(vn);
    const float sv = sinf(vn) / vn;
    mu0 = cv * mu0 + sv * v0;
    mu1 = cv * mu1 + sv * v1;
    mu2 = cv * mu2 + sv * v2;
    mu3 = cv * mu3 + sv * v3;
    float np = mu0*mu0 + mu1*mu1 + mu2*mu2 + mu3*mu3;
#pragma unroll
    for (int m = 16; m >= 1; m >>= 1) np += __shfl_xor(np, m, 32);
    const float nsc = 1.0f / fmaxf(sqrtf(np), 1e-12f);
    mu0 *= nsc; mu1 *= nsc; mu2 *= nsc; mu3 *= nsc;

    __syncthreads();
    *(v4f*)&s_mu[4 * lane] = (v4f){mu0, mu1, mu2, mu3};
    __syncthreads();
  }

  // ---- Final loss: sum_n acos(clip(<Xn,mu>))^2, mean over groups ----
  v8f f0e = zero8, f0o = zero8, f1e = zero8, f1o = zero8;
#pragma unroll
  for (int c = 0; c < 32; c += 2) {
    const v2f ae = *(const v2f*)&s_mu[4 * c + doff];
    const v2f ao = *(const v2f*)&s_mu[4 * (c + 1) + doff];
    f0e = __builtin_amdgcn_wmma_f32_16x16x4_f32(false, ae, false, xb0[c],
                                                (short)0, f0e, false, false);
    f1e = __builtin_amdgcn_wmma_f32_16x16x4_f32(false, ae, false, xb1[c],
                                                (short)0, f1e, false, false);
    f0o = __builtin_amdgcn_wmma_f32_16x16x4_f32(false, ao, false, xb0[c + 1],
                                                (short)0, f0o, false, false);
    f1o = __builtin_amdgcn_wmma_f32_16x16x4_f32(false, ao, false, xb1[c + 1],
                                                (short)0, f1o, false, false);
  }
  const float dfin = (lane < 16) ? (f0e[0] + f0o[0]) : (f1e[0] + f1o[0]);
  const float dclf = fminf(fmaxf(dfin, -CLIPF), CLIPF);
  const float th   = acosf(dclf);
  float t = th * th;
#pragma unroll
  for (int m = 16; m >= 1; m >>= 1) t += __shfl_xor(t, m, 32);
  if (lane == 0) atomicAdd(out, t * inv_groups);
}

extern "C" void kernel_launch(void* const* d_in, const int* in_sizes, int n_in,
                              void* d_out, int out_size, void* d_ws, size_t ws_size,
                              hipStream_t stream) {
  (void)n_in; (void)d_ws; (void)ws_size;
  const float* X = (const float*)d_in[0];
  float* out = (float*)d_out;
  const int total  = in_sizes[0];            // NC*K*N*D = 1000*8*32*128
  const int groups = total / (KN * KD);      // 8000
  EKCL_karcher_zero<<<1, 32, 0, stream>>>(out, out_size);
  EKCL_karcher_wmma<<<groups, 32, 0, stream>>>(X, out, 1.0f / (float)groups);
}